// interpolation_layer_5995774345370
// MI455X (gfx1250) — compile-verified
//
#include <hip/hip_runtime.h>
#include <hip/hip_bf16.h>

// Bilinear gather via TDM-staged LDS planes (MI455X / gfx1250).
//
// fm (32,256,64,64) f32, anchors (32,2048,2) f32 -> out (32,2048,256) f32.
//
//  - Direct gathering is line-request bound (each corner fetch uses 8B of a
//    unique 128B line at 16KB channel stride). Instead exploit CDNA5's 320KB
//    WGP LDS: each workgroup stages 8 full 16KB channel planes (128KB).
//  - Phase 1: TENSOR_LOAD_TO_LDS (Tensor Data Mover) DMAs the contiguous
//    128KB tile memory->LDS with ONE instruction from one wave: no VGPR
//    round trip, tracked by TENSORcnt. (Fallback: manual b128 copy.)
//  - Phase 2: 2048 anchors x 8 channels gathered from LDS (ds_load pairs),
//    bilerp in VALU, 32B non-temporal output stores (write-once stream; keeps
//    the L2-resident feature map unpolluted).
//  - Branchless clamped-origin bilerp: x0 = min(floor(px),62), tx = px-x0
//    reproduces the reference floor/ceil blend for all cases incl. px==63.

typedef float floatx4 __attribute__((ext_vector_type(4)));
typedef unsigned int v4u __attribute__((ext_vector_type(4)));
typedef int v4i __attribute__((ext_vector_type(4)));
typedef int v8i __attribute__((ext_vector_type(8)));

#define BB  32
#define CC  256
#define HH  64
#define NN  2048
#define HW  (HH * HH)          // 4096
#define CHUNK 8                // channels staged per block
#define THREADS 256
#define LDS_BYTES (CHUNK * HW * 4)   // 131072

__global__ __launch_bounds__(THREADS) void bilinear_gather_tdm_kernel(
    const float* __restrict__ fm,      // (B, C, H, W)
    const float* __restrict__ anchors, // (B, N, 2)
    float* __restrict__ out)           // (B, N, C)
{
    extern __shared__ float smem[];    // CHUNK * HW floats = 128 KB

    const int t     = threadIdx.x;
    const int b     = blockIdx.x >> 5;        // 32 chunks per batch
    const int chunk = blockIdx.x & 31;
    const int c0    = chunk * CHUNK;

    const float* __restrict__ src =
        fm + ((size_t)b * CC + c0) * (size_t)HW;   // contiguous 128KB tile

    // ---- Phase 1: stage 8 channel planes into LDS ----
#if __has_builtin(__builtin_amdgcn_tensor_load_to_lds)
    if (t < 32) {  // wave 0 only: TDM ignores EXEC; one DMA per workgroup
        // LDS offset = low 32 bits of the shared-aperture flat address
        const unsigned lds_addr = (unsigned)(uintptr_t)smem;
        const unsigned long long ga = (unsigned long long)(uintptr_t)src;

        // D# group 0: count=1 | lds_addr | global_addr[56:0] | type=2
        v4u g0;
        g0[0] = 1u;                                        // count=1, user desc
        g0[1] = lds_addr;                                  // [63:32]
        g0[2] = (unsigned)(ga & 0xFFFFFFFFull);            // [95:64]
        g0[3] = (unsigned)((ga >> 32) & 0x01FFFFFFull)     // [120:96]
              | (2u << 30);                                // type=2 ("image")

        // D# group 1: 2D tile -> 8 rows x 4096 elems, 4B elements
        const unsigned td0 = HW;     // tensor_dim0 = 4096
        const unsigned td1 = CHUNK;  // tensor_dim1 = 8
        v8i g1;
        g1[0] = (int)(2u << 16);                 // wg_mask=0, data_size=2 (4B)
        g1[1] = (int)((td0 & 0xFFFFu) << 16);    // barrier_addr=0 | dim0[15:0]
        g1[2] = (int)((td0 >> 16) | ((td1 & 0xFFFFu) << 16)); // dim0[31:16] | dim1[15:0]
        g1[3] = (int)((td1 >> 16) | ((unsigned)HW << 16));    // dim1[31:16] | tile_dim0
        g1[4] = (int)CHUNK;                      // tile_dim1=8 | tile_dim2=0
        g1[5] = (int)HW;                         // tensor_dim0_stride[31:0] = 4096
        g1[6] = 0;                               // stride[47:32] | dim1_stride[15:0]
        g1[7] = 0;                               // dim1_stride[47:16]

        v4i gz = {0, 0, 0, 0};                   // groups 2/3 unused (2D tensor)

#if defined(__clang_major__) && (__clang_major__ >= 23)
        v8i gz8 = {0, 0, 0, 0, 0, 0, 0, 0};
        __builtin_amdgcn_tensor_load_to_lds(g0, g1, gz, gz, gz8, 0);
#else
        __builtin_amdgcn_tensor_load_to_lds(g0, g1, gz, gz, 0);
#endif
        __builtin_amdgcn_s_wait_tensorcnt(0);
    }
#else
    // Fallback: cooperative b128 copy (coalesced; 32 float4 per thread)
    {
        const floatx4* __restrict__ src4 = reinterpret_cast<const floatx4*>(src);
        floatx4* dst4 = reinterpret_cast<floatx4*>(smem);
#pragma unroll
        for (int k = 0; k < (CHUNK * HW / 4) / THREADS; ++k)
            dst4[t + k * THREADS] = src4[t + k * THREADS];
    }
#endif
    __syncthreads();

    // ---- Phase 2: each thread handles 8 anchors x 8 channels ----
#pragma unroll 1
    for (int k = 0; k < NN / THREADS; ++k) {
        const int n = (k << 8) + t;   // consecutive threads -> consecutive anchors
        const float2 pa =
            *reinterpret_cast<const float2*>(anchors + ((size_t)b * NN + n) * 2);

        // scale to pixel coords and clamp (reference semantics)
        const float px = fminf(fmaxf(pa.x * (float)(HH - 1), 0.0f), (float)(HH - 1));
        const float py = fminf(fmaxf(pa.y * (float)(HH - 1), 0.0f), (float)(HH - 1));

        const int x0 = min((int)floorf(px), HH - 2);
        const int y0 = min((int)floorf(py), HH - 2);
        const float tx = px - (float)x0;   // == dx normally; 1.0 iff px == 63.0
        const float ty = py - (float)y0;   // == dy normally; 1.0 iff py == 63.0

        const int iTop = y0 * HH + x0;

        float r[CHUNK];
#pragma unroll
        for (int c = 0; c < CHUNK; ++c) {
            const float* __restrict__ p = smem + c * HW + iTop;
            const float lt = p[0];
            const float rt = p[1];
            const float lb = p[HH];
            const float rb = p[HH + 1];
            const float top = fmaf(rt - lt, tx, lt);
            const float bot = fmaf(rb - lb, tx, lb);
            r[c] = fmaf(bot - top, ty, top);
        }

        // 32B of this anchor's output: two 16B non-temporal stores
        float* o = out + ((size_t)b * NN + n) * CC + c0;  // 32B aligned
        floatx4 lo = { r[0], r[1], r[2], r[3] };
        floatx4 hi = { r[4], r[5], r[6], r[7] };
        __builtin_nontemporal_store(lo, reinterpret_cast<floatx4*>(o));
        __builtin_nontemporal_store(hi, reinterpret_cast<floatx4*>(o + 4));
    }
}

extern "C" void kernel_launch(void* const* d_in, const int* in_sizes, int n_in,
                              void* d_out, int out_size, void* d_ws, size_t ws_size,
                              hipStream_t stream) {
    const float* fm      = (const float*)d_in[0]; // (32,256,64,64)
    const float* anchors = (const float*)d_in[1]; // (32,2048,2)
    float* out           = (float*)d_out;         // (32,2048,256)

    const int blocks = BB * (CC / CHUNK);   // 32 * 32 = 1024
    bilinear_gather_tdm_kernel<<<blocks, THREADS, LDS_BYTES, stream>>>(fm, anchors, out);
}